// SimpleTransformer_55284819034459
// MI455X (gfx1250) — compile-verified
//
#include <hip/hip_runtime.h>

// ---------------------------------------------------------------------------
// Problem constants (from reference): B=4, T=512, D=1024, V=50257,
// K_CLUSTERS=225, MAX_CS=224, TOPK=1.
// ---------------------------------------------------------------------------
#define N_TOK    2048       // B*T
#define DIM      1024
#define KC       225
#define KC_TILES 15         // ceil(225/16) -> 240 padded
#define MAX_CS   224
#define NTILES_W 14         // 224/16
#define VOCAB    50257

typedef __attribute__((ext_vector_type(2))) float v2f;
typedef __attribute__((ext_vector_type(4))) float v4f;
typedef __attribute__((ext_vector_type(8))) float v8f;

// D = A(16x4, f32) * B(4x16, f32) + C(16x16, f32), full fp32 precision.
__device__ __forceinline__ v8f wmma_f32(v2f a, v2f b, v8f c) {
  return __builtin_amdgcn_wmma_f32_16x16x4_f32(
      /*neg_a=*/false, a, /*neg_b=*/false, b,
      /*c_mod=*/(short)0, c, /*reuse_a=*/false, /*reuse_b=*/false);
}

__device__ __forceinline__ v2f ldv2(const float* p) { return *(const v2f*)p; }

// ---------------------------------------------------------------------------
// Kernel 1: fill d_out with -inf. 412 MB of pure stores -> the runtime
// bottleneck (~18us at 23.3 TB/s). 128-bit non-temporal stores: the output is
// 2x the 192 MB L2, keep the stream out of the cache.
// ---------------------------------------------------------------------------
__global__ void __launch_bounds__(256)
fill_neginf_kernel(float* __restrict__ out, long n) {
  const float ninf = -__builtin_inff();
  long n4 = n >> 2;
  v4f* __restrict__ out4 = (v4f*)out;
  v4f v = {ninf, ninf, ninf, ninf};
  long i = (long)blockIdx.x * blockDim.x + threadIdx.x;
  long stride = (long)gridDim.x * blockDim.x;
  for (; i < n4; i += stride)
    __builtin_nontemporal_store(v, &out4[i]);
  if (blockIdx.x == 0 && threadIdx.x == 0)          // tail (none for this shape)
    for (long t = n4 << 2; t < n; ++t) out[t] = ninf;
}

// ---------------------------------------------------------------------------
// Kernel 2: per-token argmax over 225 centroid scores.
// One wave per 16-token tile (grid.x = 128). 15 cluster tiles of 16 (225
// padded to 240; padded columns masked to -inf before the max). K-loop of
// 256 chunks of K=4 via V_WMMA_F32_16X16X4_F32 with two accumulators.
// A layout: lanes 0-15 hold M=lane, K = k,k+1; lanes 16-31 hold K = k+2,k+3.
// B layout (symmetric): VGPR j, lanes 0-15 = row K=j, lanes 16-31 = K=j+2,
// so both operands are a single b64 load at (k + 2*half).
// C layout: element i = rows M=i (lanes 0-15) / M=i+8 (lanes 16-31), N=lane&15.
// ---------------------------------------------------------------------------
__global__ void
__attribute__((amdgpu_flat_work_group_size(32, 32), amdgpu_waves_per_eu(1)))
cluster_argmax_kernel(const float* __restrict__ x,
                      const float* __restrict__ centroids,
                      int* __restrict__ token_cluster) {
  const int lane   = threadIdx.x;     // 0..31, one wave32 per block
  const int half2  = (lane >> 4) << 1;
  const int lcol   = lane & 15;
  const int token0 = blockIdx.x * 16;

  const float* __restrict__ xrow = x + (size_t)(token0 + lcol) * DIM + half2;

  float best[8];
  int   bestc[8];
#pragma unroll
  for (int i = 0; i < 8; ++i) { best[i] = -__builtin_inff(); bestc[i] = 0; }

  for (int tile = 0; tile < KC_TILES; ++tile) {
    const int cid   = tile * 16 + lcol;             // cluster this lane feeds
    const int cclmp = cid < KC ? cid : (KC - 1);    // keep padded cols in-bounds
    const float* __restrict__ crow = centroids + (size_t)cclmp * DIM + half2;

    v8f acc0 = {};
    v8f acc1 = {};
#pragma unroll 4
    for (int k = 0; k < DIM; k += 8) {
      v2f a0 = ldv2(xrow + k);
      v2f b0 = ldv2(crow + k);
      v2f a1 = ldv2(xrow + k + 4);
      v2f b1 = ldv2(crow + k + 4);
      acc0 = wmma_f32(a0, b0, acc0);
      acc1 = wmma_f32(a1, b1, acc1);
    }

    const bool valid = cid < KC;
#pragma unroll
    for (int i = 0; i < 8; ++i) {
      float s = valid ? (acc0[i] + acc1[i]) : -__builtin_inff();
      if (s > best[i] || (s == best[i] && cid < bestc[i])) {
        best[i] = s; bestc[i] = cid;
      }
    }
  }

  // Reduce across the 16 lanes of each half (xor masks < 16 stay in a half).
#pragma unroll
  for (int m = 1; m <= 8; m <<= 1) {
#pragma unroll
    for (int i = 0; i < 8; ++i) {
      float os = __shfl_xor(best[i], m, 32);
      int   oc = __shfl_xor(bestc[i], m, 32);
      if (os > best[i] || (os == best[i] && oc < bestc[i])) {
        best[i] = os; bestc[i] = oc;
      }
    }
  }

  // lane 0 holds tokens token0+0..7 (element i), lane 16 holds token0+8..15.
  if (lcol == 0) {
#pragma unroll
    for (int i = 0; i < 8; ++i)
      token_cluster[token0 + half2 * 4 + i] = bestc[i];
  }
}

// ---------------------------------------------------------------------------
// Kernel 3: sub-logits + scatter. Tokens route to distinct clusters so they
// cannot share a B matrix; broadcast the token's x chunk into all 16 A rows
// (16x WMMA waste on ~1 GFLOP -> irrelevant vs the 412 MB store stream).
// Grid: (token, n_tile) = (2048, 14); one wave per block.
// D row M=0 (element 0, lanes 0-15) carries sub_logit[ntile*16 + lane].
// ---------------------------------------------------------------------------
__global__ void
__attribute__((amdgpu_flat_work_group_size(32, 32), amdgpu_waves_per_eu(1)))
sublogits_scatter_kernel(const float* __restrict__ x,
                         const float* __restrict__ ow,
                         const int* __restrict__ members,
                         const int* __restrict__ token_cluster,
                         float* __restrict__ out) {
  const int token = blockIdx.x;
  const int ntile = blockIdx.y;       // 0..13
  const int lane  = threadIdx.x;
  const int half2 = (lane >> 4) << 1;
  const int lcol  = lane & 15;

  const int c = token_cluster[token];                 // uniform -> scalar path
  const float* __restrict__ xt = x + (size_t)token * DIM + half2;
  const float* __restrict__ w  =
      ow + ((size_t)c * MAX_CS + (size_t)ntile * 16 + lcol) * DIM + half2;

  v8f acc0 = {};
  v8f acc1 = {};
#pragma unroll 4
  for (int k = 0; k < DIM; k += 8) {
    v2f a0 = ldv2(xt + k);            // broadcast: all 16 A rows identical
    v2f b0 = ldv2(w + k);
    v2f a1 = ldv2(xt + k + 4);
    v2f b1 = ldv2(w + k + 4);
    acc0 = wmma_f32(a0, b0, acc0);
    acc1 = wmma_f32(a1, b1, acc1);
  }

  if (half2 == 0) {
    const int m = ntile * 16 + lcol;                  // < 224 always
    const int member = members[c * MAX_CS + m];
    if (member >= 0) {
      float sub = acc0[0] + acc1[0];                  // row M=0, N=lcol
      out[(size_t)token * VOCAB + member] = sub;
    }
  }
}

// ---------------------------------------------------------------------------
// Host launcher. Inputs (setup_inputs order):
//   d_in[0] x                (2048*1024)    f32
//   d_in[1] centroids        (225*1024)     f32
//   d_in[2] organized_weight (225*224*1024) f32
//   d_in[3] cluster_members  (225*224)      i32
//   d_in[4] k                (1)            i32 (== 1, ignored)
// d_out: 2048*50257 f32. d_ws: token->cluster ids (2048 * i32), rewritten
// fully every call (deterministic, no cross-call state).
// ---------------------------------------------------------------------------
extern "C" void kernel_launch(void* const* d_in, const int* in_sizes, int n_in,
                              void* d_out, int out_size, void* d_ws, size_t ws_size,
                              hipStream_t stream) {
  const float* x         = (const float*)d_in[0];
  const float* centroids = (const float*)d_in[1];
  const float* ow        = (const float*)d_in[2];
  const int*   members   = (const int*)d_in[3];
  float*       out       = (float*)d_out;
  int*         token_cluster = (int*)d_ws;

  // 1) -inf canvas: bandwidth-bound, ~412 MB of NT stores.
  fill_neginf_kernel<<<2048, 256, 0, stream>>>(out, (long)out_size);

  // 2) routing: 128 blocks x 1 wave, 16 tokens each.
  cluster_argmax_kernel<<<N_TOK / 16, 32, 0, stream>>>(x, centroids, token_cluster);

  // 3) sub-logits + scatter: (2048 tokens, 14 weight tiles) x 1 wave.
  sublogits_scatter_kernel<<<dim3(N_TOK, NTILES_W), 32, 0, stream>>>(
      x, ow, members, token_cluster, out);
}